// Generator_52286931861627
// MI455X (gfx1250) — compile-verified
//
#include <hip/hip_runtime.h>

// GraphConv x2 on MI455X (gfx1250, wave32).
// Algebraic fusion: since IN_F=OUT_F=1, both graph aggregations reduce to
// scalar segment-sums; the only dense work is H = relu([agg1,x,1]·[u;v;b1])
// ([N,2+1]x[3,128]) and two 128-wide dots per node, which we run on the
// matrix pipe via V_WMMA_F32_16X16X4_F32 (exact f32 FMA semantics).

typedef __attribute__((ext_vector_type(2))) float v2f;
typedef __attribute__((ext_vector_type(8))) float v8f;

__global__ void zero2_kernel(float* __restrict__ a, float* __restrict__ b, int n) {
  int i = blockIdx.x * blockDim.x + threadIdx.x;
  if (i < n) { a[i] = 0.0f; b[i] = 0.0f; }
}

// Scalar segment-sum over edges: acc[dst[e]] += val[src[e]]
__global__ void edge_scatter_kernel(const long long* __restrict__ src,
                                    const long long* __restrict__ dst,
                                    const float* __restrict__ val,
                                    float* __restrict__ acc, int E) {
  int e = blockIdx.x * blockDim.x + threadIdx.x;
  if (e < E) {
    int s = (int)src[e];
    int d = (int)dst[e];
    atomicAdd(&acc[d], val[s]);
  }
}

// One wave (32 lanes) per 16-node tile.
// A (16x4 f32): lanes 0-15 hold (agg1[m], x[m]) for K=0,1; lanes 16-31 hold (1,0) for K=2,3.
// B_b (4x16 f32): lanes 0-15 hold (u[col], v[col]); lanes 16-31 hold (b1[col], 0).
// C_b = A x B_b is the H tile for feature-column block b (lane -> col, VGPR -> node row).
__global__ void node_wmma_kernel(const float* __restrict__ x,
                                 const float* __restrict__ agg1,
                                 const float* __restrict__ w1rel,   // u[128]
                                 const float* __restrict__ b1,      // b1[128]
                                 const float* __restrict__ w1root,  // v[128]
                                 const float* __restrict__ w2rel,   // [128]
                                 const float* __restrict__ w2root,  // [128]
                                 float* __restrict__ p,
                                 float* __restrict__ q,
                                 int nNodes, int nTiles) {
  const int wave = threadIdx.x >> 5;
  const int lane = threadIdx.x & 31;
  const int tile = blockIdx.x * (blockDim.x >> 5) + wave;
  if (tile >= nTiles) return;               // wave-uniform: EXEC stays all-1s for WMMA

  const int m    = lane & 15;
  const int half = lane >> 4;
  int node = tile * 16 + m;
  if (node >= nNodes) node = nNodes - 1;    // clamp loads (N=100000 is 16-divisible anyway)

  const float aval = agg1[node];
  const float cval = x[node];
  v2f A;
  A.x = half ? 1.0f : aval;                 // K = 0 / 2
  A.y = half ? 0.0f : cval;                 // K = 1 / 3

  float accP[8], accQ[8];
#pragma unroll
  for (int r = 0; r < 8; ++r) { accP[r] = 0.0f; accQ[r] = 0.0f; }

#pragma unroll
  for (int b = 0; b < 8; ++b) {
    const int col = m + 16 * b;             // feature column this lane owns in C_b
    const float u  = w1rel[col];
    const float v  = w1root[col];
    const float bb = b1[col];
    v2f B;
    B.x = half ? bb   : u;                  // K = 0 / 2
    B.y = half ? 0.0f : v;                  // K = 1 / 3
    v8f C = {};
    C = __builtin_amdgcn_wmma_f32_16x16x4_f32(
        /*neg_a=*/false, A, /*neg_b=*/false, B,
        /*c_mod=*/(short)0, C, /*reuse_a=*/false, /*reuse_b=*/false);

    const float wr = w2rel[col];
    const float wo = w2root[col];
#pragma unroll
    for (int r = 0; r < 8; ++r) {           // C[r] = H[node r+8*half, col]
      const float h = fmaxf(C[r], 0.0f);    // ReLU of layer 1
      accP[r] += h * wr;
      accQ[r] += h * wo;
    }
  }

  // Reduce over the 16 lanes of each half (xor masks < 16 never cross halves).
#pragma unroll
  for (int off = 8; off >= 1; off >>= 1) {
#pragma unroll
    for (int r = 0; r < 8; ++r) {
      accP[r] += __shfl_xor(accP[r], off, 32);
      accQ[r] += __shfl_xor(accQ[r], off, 32);
    }
  }

  if ((lane & 15) == 0) {                   // lanes 0 and 16 write nodes 0-7 / 8-15
    const int base = tile * 16 + 8 * half;
#pragma unroll
    for (int r = 0; r < 8; ++r) {
      const int n = base + r;
      if (n < nNodes) { p[n] = accP[r]; q[n] = accQ[r]; }
    }
  }
}

__global__ void finalize_kernel(const float* __restrict__ s2,
                                const float* __restrict__ q,
                                const float* __restrict__ b2,
                                float* __restrict__ out, int n) {
  int i = blockIdx.x * blockDim.x + threadIdx.x;
  if (i < n) {
    const float e = s2[i] + b2[0] + q[i];
    out[i]     = e;                         // embeddings_before
    out[n + i] = fmaxf(e, 0.0f);            // updated_x = relu(...)
  }
}

extern "C" void kernel_launch(void* const* d_in, const int* in_sizes, int n_in,
                              void* d_out, int out_size, void* d_ws, size_t ws_size,
                              hipStream_t stream) {
  const float*     x       = (const float*)d_in[0];
  const long long* ei      = (const long long*)d_in[1];   // int64 [2, E]
  const float*     W1_rel  = (const float*)d_in[2];       // [128,1] -> u
  const float*     b1      = (const float*)d_in[3];       // [128]
  const float*     W1_root = (const float*)d_in[4];       // [128,1] -> v
  const float*     W2_rel  = (const float*)d_in[5];       // [1,128]
  const float*     b2      = (const float*)d_in[6];       // [1]
  const float*     W2_root = (const float*)d_in[7];       // [1,128]
  float*           out     = (float*)d_out;

  const int N = in_sizes[0];        // 100000 (IN_F == 1)
  const int E = in_sizes[1] / 2;    // 1600000
  const long long* src = ei;
  const long long* dst = ei + E;

  // Workspace: agg1 | p | q | s2  (4*N floats = 1.6 MB)
  float* ws   = (float*)d_ws;
  float* agg1 = ws;
  float* p    = ws + (size_t)N;
  float* q    = ws + 2 * (size_t)N;
  float* s2   = ws + 3 * (size_t)N;

  const int TB = 256;

  zero2_kernel<<<(N + TB - 1) / TB, TB, 0, stream>>>(agg1, s2, N);

  edge_scatter_kernel<<<(E + TB - 1) / TB, TB, 0, stream>>>(src, dst, x, agg1, E);

  const int nTiles = (N + 15) / 16;
  const int wavesPerBlock = 8;      // 256 threads = 8 wave32s
  node_wmma_kernel<<<(nTiles + wavesPerBlock - 1) / wavesPerBlock,
                     32 * wavesPerBlock, 0, stream>>>(
      x, agg1, W1_rel, b1, W1_root, W2_rel, W2_root, p, q, N, nTiles);

  edge_scatter_kernel<<<(E + TB - 1) / TB, TB, 0, stream>>>(src, dst, p, s2, E);

  finalize_kernel<<<(N + TB - 1) / TB, TB, 0, stream>>>(s2, q, b2, out, N);
}